// Hippocampus_81260781240452
// MI455X (gfx1250) — compile-verified
//
#include <hip/hip_runtime.h>
#include <hip/hip_bf16.h>
#include <math.h>

typedef __attribute__((ext_vector_type(2))) float v2f;
typedef __attribute__((ext_vector_type(8))) float v8f;

#define BB 512
#define SS 8
#define UU 512
#define DD 32
#define GG 8
#define DIM_IN 4096
#define DAG 2048
#define K1 (DIM_IN / SS) /* 512 */
#define N1 (DAG / SS)    /* 256 */
#define K2 N1            /* 256 */
#define N2 (UU * DD / SS)/* 2048 */
#define UD (UU * DD)     /* 16384 */

#define HAS_ASYNC_LDS (__has_builtin(__builtin_amdgcn_global_load_async_to_lds_b32) && \
                       __has_builtin(__builtin_amdgcn_s_wait_asynccnt))

typedef __attribute__((address_space(1))) int* gptr_i32;
typedef __attribute__((address_space(3))) int* lptr_i32;

// fp32 WMMA: D(16x16,f32) = A(16x4,f32) * B(4x16,f32) + C
__device__ __forceinline__ v8f wmma_f32(v2f a, v2f b, v8f c) {
    return __builtin_amdgcn_wmma_f32_16x16x4_f32(
        /*neg_a=*/false, a, /*neg_b=*/false, b,
        /*c_mod=*/(short)0, c, /*reuse_a=*/false, /*reuse_b=*/false);
}

// ---------------- GEMM1: ga[b][s*N1+n] = sum_k x[b][s*K1+k] * w1[s][k][n] + b1[s][n]
__global__ __launch_bounds__(128) void k_gemm1(const float* __restrict__ x,
                                               const float* __restrict__ w1,
                                               const float* __restrict__ b1,
                                               float* __restrict__ ga) {
    const int lane = threadIdx.x & 31;
    const int wave = threadIdx.x >> 5;
    const int s  = blockIdx.z;
    const int m0 = blockIdx.y * 16;
    const int n0 = blockIdx.x * 64 + wave * 16;
    const int r  = lane & 15;
    const int kh = lane >> 4;   // K pair base = 2*kh

    const float* ap = x  + (size_t)(m0 + r) * DIM_IN + s * K1 + 2 * kh;
    const float* bp = w1 + ((size_t)s * K1 + 2 * kh) * N1 + n0 + r;
    v8f acc = {};
    for (int k = 0; k < K1; k += 4) {
        v2f a, b;
        a.x = ap[k];              a.y = ap[k + 1];
        b.x = bp[(size_t)k * N1]; b.y = bp[(size_t)(k + 1) * N1];
        acc = wmma_f32(a, b, acc);
    }
    const int   cn   = n0 + r;
    const float bias = b1[s * N1 + cn];
    float* out = ga + (size_t)(m0 + kh * 8) * DAG + s * N1 + cn;
#pragma unroll
    for (int v = 0; v < 8; ++v) out[(size_t)v * DAG] = acc[v] + bias;
}

// ---------------- GEMM2: att[b][s*N2+n] = sum_k ga[b][s*K2+k] * w2[s][k][n] + b2[s][n]
__global__ __launch_bounds__(128) void k_gemm2(const float* __restrict__ ga,
                                               const float* __restrict__ w2,
                                               const float* __restrict__ b2,
                                               float* __restrict__ att) {
    const int lane = threadIdx.x & 31;
    const int wave = threadIdx.x >> 5;
    const int s  = blockIdx.z;
    const int m0 = blockIdx.y * 16;
    const int n0 = blockIdx.x * 64 + wave * 16;
    const int r  = lane & 15;
    const int kh = lane >> 4;

    const float* ap = ga + (size_t)(m0 + r) * DAG + s * K2 + 2 * kh;
    const float* bp = w2 + ((size_t)s * K2 + 2 * kh) * N2 + n0 + r;
    v8f acc = {};
    for (int k = 0; k < K2; k += 4) {
        v2f a, b;
        a.x = ap[k];              a.y = ap[k + 1];
        b.x = bp[(size_t)k * N2]; b.y = bp[(size_t)(k + 1) * N2];
        acc = wmma_f32(a, b, acc);
    }
    const int   cn   = n0 + r;
    const float bias = b2[s * N2 + cn];
    float* out = att + (size_t)(m0 + kh * 8) * UD + s * N2 + cn;
#pragma unroll
    for (int v = 0; v < 8; ++v) out[(size_t)v * UD] = acc[v] + bias;
}

// ---------------- weights raw: w[b][m][u] = mask(temp[u] * sum_d att[b][u][d]*memA[m][u][d])
__global__ __launch_bounds__(256) void k_weights(const float* __restrict__ att,
                                                 const float* __restrict__ memA,
                                                 const float* __restrict__ randu,
                                                 const float* __restrict__ temperature,
                                                 float* __restrict__ w) {
    const int lane = threadIdx.x & 31;
    const int wave = threadIdx.x >> 5;
    const int u  = blockIdx.x * 8 + wave;
    const int b0 = blockIdx.y * 16;
    const int m0 = blockIdx.z * 16;
    const int r  = lane & 15;
    const int kh = lane >> 4;

    // Prefetch the scattered rand_u mask lines; they are consumed only in the
    // epilogue, so this overlaps their fetch with the WMMA loop.
    const int m = m0 + r;
#pragma unroll
    for (int v = 0; v < 8; ++v) {
        const int b = b0 + v + kh * 8;
        __builtin_prefetch(&randu[((size_t)b * BB + m) * UU + u], 0, 1);
    }

    const float* ap = att  + (size_t)(b0 + r) * UD + u * DD + 2 * kh; // A[M=b][K=d]
    const float* bp = memA + (size_t)(m0 + r) * UD + u * DD + 2 * kh; // B[K=d][N=m]
    v8f acc = {};
#pragma unroll
    for (int k = 0; k < DD; k += 4) {
        v2f a, b;
        a.x = ap[k]; a.y = ap[k + 1];
        b.x = bp[k]; b.y = bp[k + 1];
        acc = wmma_f32(a, b, acc);
    }
    const float t   = temperature[u];
    const float NEG = -__builtin_inff();
#pragma unroll
    for (int v = 0; v < 8; ++v) {
        const int    b   = b0 + v + kh * 8;
        const size_t idx = ((size_t)b * BB + m) * UU + u;
        float val = acc[v] * t;
        if (randu[idx] < 0.1f || b == m) val = NEG;
        w[idx] = val;
    }
}

// ---------------- group softmax over m = g + 8*i (i in [0,64)), in place, /8
__global__ __launch_bounds__(128) void k_softmax(float* __restrict__ w) {
    __shared__ float tile[64][128];
    const int t  = threadIdx.x;
    const int b  = blockIdx.z;
    const int g  = blockIdx.y;
    const int u0 = blockIdx.x * 128;
    float* base = w + ((size_t)b * BB + g) * UU + u0;
#if HAS_ASYNC_LDS
    // gfx1250 async copy: memory -> LDS directly, tracked by ASYNCcnt.
#pragma unroll 8
    for (int i = 0; i < 64; ++i) {
        __builtin_amdgcn_global_load_async_to_lds_b32(
            (gptr_i32)(base + (size_t)i * 8 * UU + t),
            (lptr_i32)&tile[i][t],
            0, 0);
    }
    __builtin_amdgcn_s_wait_asynccnt(0);
    __syncthreads();
#else
#pragma unroll 4
    for (int i = 0; i < 64; ++i) tile[i][t] = base[(size_t)i * 8 * UU + t];
    __syncthreads();
#endif
    float mx = -__builtin_inff();
    for (int i = 0; i < 64; ++i) mx = fmaxf(mx, tile[i][t]);
    float sum = 0.f;
    for (int i = 0; i < 64; ++i) sum += __expf(tile[i][t] - mx);
    const float inv = 1.0f / (8.0f * sum);
    float* outp = base + t;
    for (int i = 0; i < 64; ++i)
        outp[(size_t)i * 8 * UU] = __expf(tile[i][t] - mx) * inv;
}

// ---------------- outputs[b][u][d] = sum_m w[b][m][u] * memO[m][u][d]
__global__ __launch_bounds__(256) void k_outputs(const float* __restrict__ w,
                                                 const float* __restrict__ memO,
                                                 float* __restrict__ out) {
    const int lane = threadIdx.x & 31;
    const int wave = threadIdx.x >> 5;
    const int u  = blockIdx.x * 8 + wave;
    const int b0 = blockIdx.y * 16;
    const int r  = lane & 15;
    const int kh = lane >> 4;

    const float* ap = w + ((size_t)(b0 + r) * BB + 2 * kh) * UU + u; // A[M=b][K=m]
    const float* bp = memO + (size_t)(2 * kh) * UD + u * DD + r;     // B[K=m][N=d]
    v8f acc0 = {}, acc1 = {};
    for (int k = 0; k < BB; k += 4) {
        v2f a, bv0, bv1;
        a.x = ap[(size_t)k * UU];
        a.y = ap[(size_t)(k + 1) * UU];
        const float* bk = bp + (size_t)k * UD;
        bv0.x = bk[0];  bv0.y = bk[UD];
        bv1.x = bk[16]; bv1.y = bk[UD + 16];
        acc0 = wmma_f32(a, bv0, acc0);
        acc1 = wmma_f32(a, bv1, acc1);
    }
    float* op = out + (size_t)(b0 + kh * 8) * UD + u * DD + r;
#pragma unroll
    for (int v = 0; v < 8; ++v) {
        op[(size_t)v * UD]      = acc0[v];
        op[(size_t)v * UD + 16] = acc1[v];
    }
}

extern "C" void kernel_launch(void* const* d_in, const int* in_sizes, int n_in,
                              void* d_out, int out_size, void* d_ws, size_t ws_size,
                              hipStream_t stream) {
    const float* x    = (const float*)d_in[0];
    const float* memA = (const float*)d_in[1];
    const float* memO = (const float*)d_in[2];
    const float* ru   = (const float*)d_in[3];
    const float* w1   = (const float*)d_in[4];
    const float* b1   = (const float*)d_in[5];
    const float* w2   = (const float*)d_in[6];
    const float* b2   = (const float*)d_in[7];
    const float* temp = (const float*)d_in[8];

    float* outF = (float*)d_out;
    float* att  = outF;                 // [B,U,D]   = 8,388,608
    float* ga   = outF + 8388608;       // [B,DAG]   = 1,048,576
    float* wbuf = outF + 9437184;       // [B,B,U]   = 134,217,728
    float* outs = outF + 143654912;     // [B,U,D]   = 8,388,608

    k_gemm1  <<<dim3(N1 / 64, BB / 16, SS),      128, 0, stream>>>(x, w1, b1, ga);
    k_gemm2  <<<dim3(N2 / 64, BB / 16, SS),      128, 0, stream>>>(ga, w2, b2, att);
    k_weights<<<dim3(UU / 8, BB / 16, BB / 16),  256, 0, stream>>>(att, memA, ru, temp, wbuf);
    k_softmax<<<dim3(UU / 128, GG, BB),          128, 0, stream>>>(wbuf);
    k_outputs<<<dim3(UU / 8, BB / 16),           256, 0, stream>>>(wbuf, memO, outs);
}